// GAT_13821204758764
// MI455X (gfx1250) — compile-verified
//
#include <hip/hip_runtime.h>
#include <hip/hip_bf16.h>

#define NEG_SLOPE 0.2f

typedef float v2f __attribute__((ext_vector_type(2)));
typedef float v8f __attribute__((ext_vector_type(8)));

// ---------------------------------------------------------------------------
// GEMM: C[M,256] = A[M,256] * B[256,256], fp32, V_WMMA_F32_16X16X4_F32.
// One block = 256 threads = 8 waves; block owns one 16-row M tile staged in
// LDS; each wave computes two 16-col N tiles (8 waves * 2 = 16 = 256/16).
// ---------------------------------------------------------------------------
__global__ __launch_bounds__(256) void gemm256_wmma(const float* __restrict__ A,
                                                    const float* __restrict__ B,
                                                    float* __restrict__ C) {
  __shared__ float As[16 * 260];  // pad 260 -> conflict-free b64 reads
  const int tid = threadIdx.x;
  const int m0 = blockIdx.x * 16;

  // Stage 16x256 A tile (coalesced float4 loads, aligned LDS float4 stores).
  const float4* Ag = (const float4*)(A + (size_t)m0 * 256);
#pragma unroll
  for (int j = 0; j < 4; ++j) {
    int q = tid + 256 * j;        // float4 index within tile: 0..1023
    int r = q >> 6;               // row 0..15
    int c4 = q & 63;              // float4 column
    float4 v = Ag[(size_t)r * 64 + c4];
    *(float4*)(&As[r * 260 + c4 * 4]) = v;
  }
  __syncthreads();

  const int lane = tid & 31;
  const int wv = tid >> 5;        // 0..7
  const int ml = lane & 15;       // A row / B col within tile
  const int kh = (lane >> 4) * 2; // K sub-offset: 0 (lanes 0-15) or 2 (16-31)

#pragma unroll
  for (int t = 0; t < 2; ++t) {
    const int nt = wv * 2 + t;          // N tile 0..15
    const int nc = nt * 16 + ml;        // global column
    v8f acc = {0.f, 0.f, 0.f, 0.f, 0.f, 0.f, 0.f, 0.f};
#pragma unroll 4
    for (int k = 0; k < 256; k += 4) {
      v2f a = *(const v2f*)(&As[ml * 260 + k + kh]);   // A[m][k+kh], A[m][k+kh+1]
      v2f b;
      b.x = B[(size_t)(k + kh) * 256 + nc];            // B[k+kh][n]
      b.y = B[(size_t)(k + kh + 1) * 256 + nc];        // B[k+kh+1][n]
      acc = __builtin_amdgcn_wmma_f32_16x16x4_f32(
          /*neg_a=*/false, a, /*neg_b=*/false, b,
          /*c_mod=*/(short)0, acc, /*reuse_a=*/false, /*reuse_b=*/false);
    }
    const int rbase = m0 + ((lane >> 4) * 8);  // lanes 16-31 hold rows M+8
#pragma unroll
    for (int r = 0; r < 8; ++r)
      C[(size_t)(rbase + r) * 256 + nt * 16 + ml] = acc[r];
  }
}

// ---------------------------------------------------------------------------
// Per-node attention logits: al_s[n,h] = <h[n,h,:], att_src[h,:]> (same for dst)
// ---------------------------------------------------------------------------
__global__ void attn_logits_kernel(const float* __restrict__ h,
                                   const float* __restrict__ as_,
                                   const float* __restrict__ ad_,
                                   float* __restrict__ alS, float* __restrict__ alD,
                                   int n_nodes, int heads, int ch) {
  int tid = blockIdx.x * blockDim.x + threadIdx.x;
  if (tid >= n_nodes * heads) return;
  int n = tid / heads, hd = tid - n * heads;
  const float* hp = h + (size_t)n * (heads * ch) + hd * ch;
  const float* ap = as_ + hd * ch;
  const float* bp = ad_ + hd * ch;
  float ss = 0.f, sd = 0.f;
  for (int c = 0; c < ch; ++c) { float v = hp[c]; ss += v * ap[c]; sd += v * bp[c]; }
  alS[tid] = ss;
  alD[tid] = sd;
}

// Monotonic float<->uint key for atomicMax-based segment max.
__device__ __forceinline__ unsigned fkey(float x) {
  unsigned u = __float_as_uint(x);
  return u ^ (((unsigned)((int)u >> 31)) | 0x80000000u);
}
__device__ __forceinline__ float funkey(unsigned k) {
  unsigned m = (k & 0x80000000u) ? 0x80000000u : 0xFFFFFFFFu;
  return __uint_as_float(k ^ m);
}

__device__ __forceinline__ void edge_ends(const int* __restrict__ ei, int e, int ne,
                                          int& src, int& dst) {
  if (e < ne) { src = ei[e]; dst = ei[ne + e]; }   // edge_index rows 0/1
  else        { src = dst = e - ne; }              // self loop
}

// ---------------------------------------------------------------------------
// Pass 1: segment max of leaky_relu(al_s[src]+al_d[dst]) over dst.
// ---------------------------------------------------------------------------
__global__ void edge_max_kernel(const int* __restrict__ ei, const float* __restrict__ alS,
                                const float* __restrict__ alD, unsigned* __restrict__ mkey,
                                int ne, int n_nodes, int heads) {
  int tid = blockIdx.x * blockDim.x + threadIdx.x;
  int tot = (ne + n_nodes) * heads;
  if (tid >= tot) return;
  int e = tid / heads, hd = tid - e * heads;
  int src, dst;
  edge_ends(ei, e, ne, src, dst);
  float x = alS[src * heads + hd] + alD[dst * heads + hd];
  x = (x > 0.f) ? x : NEG_SLOPE * x;
  atomicMax(&mkey[dst * heads + hd], fkey(x));
}

// ---------------------------------------------------------------------------
// Pass 2: denom[dst] = sum exp(e - m[dst]).
// ---------------------------------------------------------------------------
__global__ void edge_denom_kernel(const int* __restrict__ ei, const float* __restrict__ alS,
                                  const float* __restrict__ alD,
                                  const unsigned* __restrict__ mkey,
                                  float* __restrict__ denom,
                                  int ne, int n_nodes, int heads) {
  int tid = blockIdx.x * blockDim.x + threadIdx.x;
  int tot = (ne + n_nodes) * heads;
  if (tid >= tot) return;
  int e = tid / heads, hd = tid - e * heads;
  int src, dst;
  edge_ends(ei, e, ne, src, dst);
  float x = alS[src * heads + hd] + alD[dst * heads + hd];
  x = (x > 0.f) ? x : NEG_SLOPE * x;
  float m = funkey(mkey[dst * heads + hd]);
  atomicAdd(&denom[dst * heads + hd], expf(x - m));
}

// ---------------------------------------------------------------------------
// Pass 3: out[dst] += alpha * h[src]. One wave per edge; 32 lanes x 8 = 256 ch.
// ---------------------------------------------------------------------------
__global__ __launch_bounds__(256) void edge_agg_kernel(
    const int* __restrict__ ei, const float* __restrict__ alS,
    const float* __restrict__ alD, const unsigned* __restrict__ mkey,
    const float* __restrict__ denom, const float* __restrict__ hsrc,
    float* __restrict__ out, int ne, int n_nodes, int heads, int chshift) {
  int gw = (blockIdx.x * blockDim.x + threadIdx.x) >> 5;
  int lane = threadIdx.x & 31;
  int tot = ne + n_nodes;
  if (gw >= tot) return;
  int src, dst;
  edge_ends(ei, gw, ne, src, dst);
  float alpha[4];
  for (int hd = 0; hd < heads; ++hd) {
    float x = alS[src * heads + hd] + alD[dst * heads + hd];
    x = (x > 0.f) ? x : NEG_SLOPE * x;
    float m = funkey(mkey[dst * heads + hd]);
    alpha[hd] = expf(x - m) / denom[dst * heads + hd];
  }
  const float* hp = hsrc + (size_t)src * 256;
  float* op = out + (size_t)dst * 256;
#pragma unroll
  for (int j = 0; j < 8; ++j) {
    int c = lane + 32 * j;
    atomicAdd(&op[c], alpha[c >> chshift] * hp[c]);
  }
}

// ---------------------------------------------------------------------------
// Bias (+ optional ReLU), 256-channel rows.
// ---------------------------------------------------------------------------
__global__ void bias_act_kernel(const float* __restrict__ in, const float* __restrict__ b,
                                float* __restrict__ out, int total, int do_relu) {
  int tid = blockIdx.x * blockDim.x + threadIdx.x;
  if (tid >= total) return;
  float v = in[tid] + b[tid & 255];
  out[tid] = do_relu ? fmaxf(v, 0.f) : v;
}

extern "C" void kernel_launch(void* const* d_in, const int* in_sizes, int n_in,
                              void* d_out, int out_size, void* d_ws, size_t ws_size,
                              hipStream_t stream) {
  (void)n_in; (void)out_size; (void)ws_size;
  const float* x   = (const float*)d_in[0];
  const int*   ei  = (const int*)d_in[1];
  const float* W1  = (const float*)d_in[2];
  const float* as1 = (const float*)d_in[3];
  const float* ad1 = (const float*)d_in[4];
  const float* b1  = (const float*)d_in[5];
  const float* W2  = (const float*)d_in[6];
  const float* as2 = (const float*)d_in[7];
  const float* ad2 = (const float*)d_in[8];
  const float* b2  = (const float*)d_in[9];
  float* out = (float*)d_out;

  const int n_nodes = in_sizes[0] / 256;  // 20000
  const int ne = in_sizes[1] / 2;         // 320000 (before self loops)
  const int tot_e = ne + n_nodes;
  const size_t nodef = (size_t)n_nodes * 256;

  // Workspace layout (floats): P | H1 | alS | alD | mkey | denom  (~42 MB)
  char* ws = (char*)d_ws;
  float*    P     = (float*)ws;
  float*    H1    = (float*)(ws + nodef * sizeof(float));
  float*    alS   = (float*)(ws + 2 * nodef * sizeof(float));
  float*    alD   = alS + (size_t)n_nodes * 4;
  unsigned* mkey  = (unsigned*)(alD + (size_t)n_nodes * 4);
  float*    denom = (float*)(mkey + (size_t)n_nodes * 4);

  dim3 blk(256);
  const int mtiles = n_nodes / 16;  // 1250

  // ---------------- Layer 1 (H=4, C=64) ----------------
  gemm256_wmma<<<mtiles, blk, 0, stream>>>(x, W1, P);
  attn_logits_kernel<<<(n_nodes * 4 + 255) / 256, blk, 0, stream>>>(
      P, as1, ad1, alS, alD, n_nodes, 4, 64);
  hipMemsetAsync(mkey, 0, (size_t)n_nodes * 4 * sizeof(unsigned), stream);
  hipMemsetAsync(denom, 0, (size_t)n_nodes * 4 * sizeof(float), stream);
  hipMemsetAsync(H1, 0, nodef * sizeof(float), stream);
  edge_max_kernel<<<(tot_e * 4 + 255) / 256, blk, 0, stream>>>(
      ei, alS, alD, mkey, ne, n_nodes, 4);
  edge_denom_kernel<<<(tot_e * 4 + 255) / 256, blk, 0, stream>>>(
      ei, alS, alD, mkey, denom, ne, n_nodes, 4);
  edge_agg_kernel<<<((size_t)tot_e * 32 + 255) / 256, blk, 0, stream>>>(
      ei, alS, alD, mkey, denom, P, H1, ne, n_nodes, 4, 6);
  bias_act_kernel<<<(nodef + 255) / 256, blk, 0, stream>>>(H1, b1, H1, (int)nodef, 1);

  // ---------------- Layer 2 (H=1, C=256) ----------------
  gemm256_wmma<<<mtiles, blk, 0, stream>>>(H1, W2, P);
  attn_logits_kernel<<<(n_nodes + 255) / 256, blk, 0, stream>>>(
      P, as2, ad2, alS, alD, n_nodes, 1, 256);
  hipMemsetAsync(mkey, 0, (size_t)n_nodes * sizeof(unsigned), stream);
  hipMemsetAsync(denom, 0, (size_t)n_nodes * sizeof(float), stream);
  hipMemsetAsync(out, 0, nodef * sizeof(float), stream);
  edge_max_kernel<<<(tot_e + 255) / 256, blk, 0, stream>>>(
      ei, alS, alD, mkey, ne, n_nodes, 1);
  edge_denom_kernel<<<(tot_e + 255) / 256, blk, 0, stream>>>(
      ei, alS, alD, mkey, denom, ne, n_nodes, 1);
  edge_agg_kernel<<<((size_t)tot_e * 32 + 255) / 256, blk, 0, stream>>>(
      ei, alS, alD, mkey, denom, P, out, ne, n_nodes, 1, 8);
  bias_act_kernel<<<(nodef + 255) / 256, blk, 0, stream>>>(out, b2, out, (int)nodef, 0);
}